// MergedQKVParallelLinearWithDelta_79766132621352
// MI455X (gfx1250) — compile-verified
//
#include <hip/hip_runtime.h>

#define TOKENS   4096
#define HIDDEN   4096
#define NTOT     6144
#define QS       4096
#define KVS      1024
#define BM       128
#define BN       128
#define KS       16
#define LSTR     20      // padded float stride for A/B LDS tiles (bank-conflict free)
#define NKIT     (HIDDEN / KS)

typedef __attribute__((ext_vector_type(2))) float v2f;
typedef __attribute__((ext_vector_type(4))) float v4f;
typedef __attribute__((ext_vector_type(8))) float v8f;
typedef __attribute__((ext_vector_type(4))) int   v4i;

#if defined(__has_builtin)
#  if __has_builtin(__builtin_amdgcn_global_load_async_to_lds_b128)
#    define HAVE_ASYNC_LDS 1
#  endif
#endif
#ifndef HAVE_ASYNC_LDS
#  define HAVE_ASYNC_LDS 0
#endif

#if HAVE_ASYNC_LDS
typedef __attribute__((address_space(1))) v4i g_v4i;   // global int4
typedef __attribute__((address_space(3))) v4i l_v4i;   // LDS int4

__device__ __forceinline__ void async_b128(const void* g, void* l) {
    __builtin_amdgcn_global_load_async_to_lds_b128((g_v4i*)g, (l_v4i*)l, 0, 0);
}
__device__ __forceinline__ void wait_async0() {
#if __has_builtin(__builtin_amdgcn_s_wait_asynccnt)
    __builtin_amdgcn_s_wait_asynccnt(0);
#else
    asm volatile("s_wait_asynccnt 0" ::: "memory");
#endif
}
#endif

__global__ __launch_bounds__(256)
void qkv_delta_wmma_kernel(const float* __restrict__ x,
                           const float* __restrict__ bw,
                           const int*   __restrict__ qw_q,
                           const int*   __restrict__ qw_k,
                           const int*   __restrict__ qw_v,
                           const int*   __restrict__ qz_q,
                           const int*   __restrict__ qz_k,
                           const int*   __restrict__ qz_v,
                           const float* __restrict__ sc_q,
                           const float* __restrict__ sc_k,
                           const float* __restrict__ sc_v,
                           const int*   __restrict__ indices,
                           float*       __restrict__ out)
{
    __shared__ float lA[2][BM * LSTR];      // x tile, [row][k] padded
    __shared__ float lB[2][BN * LSTR];      // base_weight tile, [n][k] padded
    __shared__ int   lQ[2][4 * 2 * BN];     // delta qweight tile, [d][kp][n]

    const int m0 = blockIdx.y * BM;
    const int n0 = blockIdx.x * BN;

    // ---- slice select (q / k / v); BN=128 never straddles 4096 or 5120 ----
    const int*   qw;  const int* qz;  const float* sc;
    int nslice, ncol0;
    if (n0 < QS)            { qw = qw_q; qz = qz_q; sc = sc_q; nslice = QS;  ncol0 = n0; }
    else if (n0 < QS + KVS) { qw = qw_k; qz = qz_k; sc = sc_k; nslice = KVS; ncol0 = n0 - QS; }
    else                    { qw = qw_v; qz = qz_v; sc = sc_v; nslice = KVS; ncol0 = n0 - QS - KVS; }

    const int t    = threadIdx.x;
    const int lane = t & 31;
    const int w    = t >> 5;        // wave 0..7
    const int wm   = w >> 2;        // 0..1 : 64-row group
    const int wn   = w & 3;         // 0..3 : 32-col group
    const int l16  = lane & 15;
    const int h    = lane >> 4;     // lane half selects K+2 in A/B frags, M+8 in C

    // ---- per-wave per-column dequant constants: scale, -(z+1)*scale ----
    float scl[4][2], nzs[4][2];
    #pragma unroll
    for (int nt = 0; nt < 2; ++nt) {
        const int nc = ncol0 + wn * 32 + nt * 16 + l16;
        #pragma unroll
        for (int d = 0; d < 4; ++d) {
            const float s = sc[d * nslice + nc];
            const int   z = (qz[d * (nslice >> 3) + (nc >> 3)] >> ((nc & 7) * 4)) & 0xF;
            scl[d][nt] = s;
            nzs[d][nt] = -(float)(z + 1) * s;
        }
    }

    // ---- per-row delta index (A lane holds row m = base + lane%16 in both halves) ----
    int idxv[4];
    #pragma unroll
    for (int mt = 0; mt < 4; ++mt)
        idxv[mt] = indices[m0 + wm * 64 + mt * 16 + l16];

    v8f acc[4][2];
    #pragma unroll
    for (int mt = 0; mt < 4; ++mt)
        #pragma unroll
        for (int nt = 0; nt < 2; ++nt)
            acc[mt][nt] = (v8f)(0.0f);

    // ---- cooperative global->LDS load assignments (256 threads) ----
    const int ldRow = t >> 1;           // 0..127
    const int ldK   = (t & 1) * 8;      // 0 or 8
    const float* gA = x  + (size_t)(m0 + ldRow) * HIDDEN + ldK;
    const float* gB = bw + (size_t)(n0 + ldRow) * HIDDEN + ldK;
    const int qd  = t >> 6;             // delta 0..3
    const int qkp = (t >> 5) & 1;       // pack row 0..1
    const int qn  = (t & 31) * 4;       // 4 int32 columns

    // ---- compute over an LDS buffer (160 WMMA per call) ----
    auto compute_tile = [&](int buf) {
        #pragma unroll
        for (int s = 0; s < 4; ++s) {           // 4 WMMA K-steps of 4
            const int klo = s * 4;

            v2f afr[4], bfr[2];
            #pragma unroll
            for (int mt = 0; mt < 4; ++mt)
                afr[mt] = *(const v2f*)(&lA[buf][(wm * 64 + mt * 16 + l16) * LSTR + klo + 2 * h]);
            #pragma unroll
            for (int nt = 0; nt < 2; ++nt)
                bfr[nt] = *(const v2f*)(&lB[buf][(wn * 32 + nt * 16 + l16) * LSTR + klo + 2 * h]);

            // base GEMM: acc += A * B
            #pragma unroll
            for (int mt = 0; mt < 4; ++mt)
                #pragma unroll
                for (int nt = 0; nt < 2; ++nt)
                    acc[mt][nt] = __builtin_amdgcn_wmma_f32_16x16x4_f32(
                        false, afr[mt], false, bfr[nt],
                        (short)0, acc[mt][nt], false, false);

            // delta: dequant nibbles, mask A rows by delta index
            const int kp = s >> 1;      // which packed int32 row
            const int s2 = s & 1;       // which half of its 8 K-values
            #pragma unroll
            for (int d = 0; d < 4; ++d) {
                v2f db[2];
                #pragma unroll
                for (int nt = 0; nt < 2; ++nt) {
                    const int u  = lQ[buf][(d * 2 + kp) * BN + wn * 32 + nt * 16 + l16];
                    const int sh = s2 * 16 + h * 8;
                    const float q0 = (float)((u >> sh) & 0xF);
                    const float q1 = (float)((u >> (sh + 4)) & 0xF);
                    db[nt].x = fmaf(q0, scl[d][nt], nzs[d][nt]);
                    db[nt].y = fmaf(q1, scl[d][nt], nzs[d][nt]);
                }
                #pragma unroll
                for (int mt = 0; mt < 4; ++mt) {
                    v2f am;
                    am.x = (idxv[mt] == d) ? afr[mt].x : 0.0f;
                    am.y = (idxv[mt] == d) ? afr[mt].y : 0.0f;
                    #pragma unroll
                    for (int nt = 0; nt < 2; ++nt)
                        acc[mt][nt] = __builtin_amdgcn_wmma_f32_16x16x4_f32(
                            false, am, false, db[nt],
                            (short)0, acc[mt][nt], false, false);
                }
            }
        }
    };

#if HAVE_ASYNC_LDS
    // ================= async double-buffered pipeline =================
    auto issue_tile = [&](int buf, int k0) {
        async_b128(gA + k0,     &lA[buf][ldRow * LSTR + ldK]);
        async_b128(gA + k0 + 4, &lA[buf][ldRow * LSTR + ldK + 4]);
        async_b128(gB + k0,     &lB[buf][ldRow * LSTR + ldK]);
        async_b128(gB + k0 + 4, &lB[buf][ldRow * LSTR + ldK + 4]);
        async_b128(qw + (size_t)(qd * (HIDDEN / 8) + (k0 >> 3) + qkp) * nslice + ncol0 + qn,
                   &lQ[buf][(qd * 2 + qkp) * BN + qn]);
    };

    issue_tile(0, 0);
    for (int i = 0; i < NKIT; ++i) {
        const int cur = i & 1;
        wait_async0();            // own async loads for buf `cur` complete
        __syncthreads();          // everyone's loads landed; everyone done reading buf cur^1
        if (i + 1 < NKIT)
            issue_tile(cur ^ 1, (i + 1) * KS);
        compute_tile(cur);
    }
#else
    // ================= synchronous fallback =================
    for (int k0 = 0; k0 < HIDDEN; k0 += KS) {
        __syncthreads();   // previous iteration's readers done

        const v4f a0 = *(const v4f*)(gA + k0);
        const v4f a1 = *(const v4f*)(gA + k0 + 4);
        const v4f b0 = *(const v4f*)(gB + k0);
        const v4f b1 = *(const v4f*)(gB + k0 + 4);
        const v4i qv = *(const v4i*)(qw + (size_t)(qd * (HIDDEN / 8) + (k0 >> 3) + qkp) * nslice
                                        + ncol0 + qn);

        *(v4f*)(&lA[0][ldRow * LSTR + ldK    ]) = a0;
        *(v4f*)(&lA[0][ldRow * LSTR + ldK + 4]) = a1;
        *(v4f*)(&lB[0][ldRow * LSTR + ldK    ]) = b0;
        *(v4f*)(&lB[0][ldRow * LSTR + ldK + 4]) = b1;
        *(v4i*)(&lQ[0][(qd * 2 + qkp) * BN + qn]) = qv;

        __syncthreads();

        if (k0 + KS < HIDDEN) {                 // CDNA5 global_prefetch_b8
            __builtin_prefetch(gA + k0 + KS, 0, 0);
            __builtin_prefetch(gB + k0 + KS, 0, 0);
        }
        compute_tile(0);
    }
#endif

    // ---- store C: VGPR r holds M = r + 8*half, N = lane%16 ----
    #pragma unroll
    for (int mt = 0; mt < 4; ++mt) {
        #pragma unroll
        for (int nt = 0; nt < 2; ++nt) {
            const int col = n0 + wn * 32 + nt * 16 + l16;
            #pragma unroll
            for (int r = 0; r < 8; ++r) {
                const int row = m0 + wm * 64 + mt * 16 + h * 8 + r;
                out[(size_t)row * NTOT + col] = acc[mt][nt][r];
            }
        }
    }
}

extern "C" void kernel_launch(void* const* d_in, const int* in_sizes, int n_in,
                              void* d_out, int out_size, void* d_ws, size_t ws_size,
                              hipStream_t stream) {
    (void)in_sizes; (void)n_in; (void)d_ws; (void)ws_size; (void)out_size;
    const float* x    = (const float*)d_in[0];
    const float* bw   = (const float*)d_in[1];
    const int*   qwq  = (const int*)d_in[2];
    const int*   qwk  = (const int*)d_in[3];
    const int*   qwv  = (const int*)d_in[4];
    const int*   qzq  = (const int*)d_in[5];
    const int*   qzk  = (const int*)d_in[6];
    const int*   qzv  = (const int*)d_in[7];
    const float* scq  = (const float*)d_in[8];
    const float* sck  = (const float*)d_in[9];
    const float* scv  = (const float*)d_in[10];
    const int*   idx  = (const int*)d_in[11];
    float*       outp = (float*)d_out;

    dim3 grid(NTOT / BN, TOKENS / BM);   // 48 x 32
    dim3 block(256);
    qkv_delta_wmma_kernel<<<grid, block, 0, stream>>>(
        x, bw, qwq, qwk, qwv, qzq, qzk, qzv, scq, sck, scv, idx, outp);
}